// Model_57638461112679
// MI455X (gfx1250) — compile-verified
//
#include <hip/hip_runtime.h>
#include <math.h>

// ---- problem constants (n = 1024 fixed by setup_inputs) ----
#define N_PTS 1024
#define IN_F  68
#define KNN   8
#define E_F   32
#define XF    132          // IN + 2E  (x | e1 | e2)
#define FEAT2 264          // 2*XF
#define NPAIR 523776       // 1024*1023/2
#define EPS   1e-5f
#define SLOPE 0.01f

typedef __attribute__((ext_vector_type(2))) float v2f;
typedef __attribute__((ext_vector_type(8))) float v8f;

__device__ __forceinline__ float lrelu(float v) { return v >= 0.f ? v : v * SLOPE; }

// ---------------- K1: batch-norm of feat, write into xf[:, 0:68] ----------------
__global__ void k_bn(const float* __restrict__ feat, const float* __restrict__ g,
                     const float* __restrict__ b, float* __restrict__ xf) {
  __shared__ float red[128];
  const int f = blockIdx.x, tid = threadIdx.x;
  float s = 0.f;
  for (int r = tid; r < N_PTS; r += 128) s += feat[r * IN_F + f];
  red[tid] = s; __syncthreads();
  for (int o = 64; o > 0; o >>= 1) { if (tid < o) red[tid] += red[tid + o]; __syncthreads(); }
  const float mu = red[0] / (float)N_PTS;
  __syncthreads();
  float s2 = 0.f;
  for (int r = tid; r < N_PTS; r += 128) { float d = feat[r * IN_F + f] - mu; s2 += d * d; }
  red[tid] = s2; __syncthreads();
  for (int o = 64; o > 0; o >>= 1) { if (tid < o) red[tid] += red[tid + o]; __syncthreads(); }
  const float rs = rsqrtf(red[0] / (float)N_PTS + EPS);
  const float gg = g[f], bb = b[f];
  for (int r = tid; r < N_PTS; r += 128)
    xf[r * XF + f] = (feat[r * IN_F + f] - mu) * rs * gg + bb;
}

// ---------------- K2: kNN on first 4 normalized features (L1), top-8 smallest ----------------
__global__ void k_knn(const float* __restrict__ xf, int* __restrict__ idx) {
  __shared__ float dist[N_PTS];
  const int i = blockIdx.x, tid = threadIdx.x;
  const float c0 = xf[i * XF + 0], c1 = xf[i * XF + 1], c2 = xf[i * XF + 2], c3 = xf[i * XF + 3];
  for (int j = tid; j < N_PTS; j += 128) {
    dist[j] = fabsf(c0 - xf[j * XF + 0]) + fabsf(c1 - xf[j * XF + 1]) +
              fabsf(c2 - xf[j * XF + 2]) + fabsf(c3 - xf[j * XF + 3]);
  }
  __syncthreads();
  if (tid == 0) {
    float bd[KNN]; int bi[KNN];
    for (int k = 0; k < KNN; ++k) { bd[k] = 3.4e38f; bi[k] = -1; }
    for (int j = 0; j < N_PTS; ++j) {
      const float d = dist[j];
      if (d < bd[KNN - 1]) {                // strict < keeps ties on lower index (stable)
        int pos = KNN - 1;
        while (pos > 0 && d < bd[pos - 1]) { bd[pos] = bd[pos - 1]; bi[pos] = bi[pos - 1]; --pos; }
        bd[pos] = d; bi[pos] = j;
      }
    }
    for (int k = 0; k < KNN; ++k) idx[i * KNN + k] = bi[k];
  }
}

// ---------------- K3: EdgeConv1: gather(68) -> 544x32 -> lrelu -> 32x32 -> lrelu -> xf[:,68:100] ----
__global__ void k_ec1(float* __restrict__ xf, const int* __restrict__ idx,
                      const float* __restrict__ w1, const float* __restrict__ b1,
                      const float* __restrict__ w2, const float* __restrict__ b2) {
  __shared__ float nb[KNN * IN_F];
  __shared__ float t1[E_F];
  const int i = blockIdx.x, tid = threadIdx.x;
  for (int t = tid; t < KNN * IN_F; t += 64) {
    const int k = t / IN_F, f = t % IN_F;
    nb[t] = xf[idx[i * KNN + k] * XF + f];
  }
  __syncthreads();
  if (tid < E_F) {
    float acc = b1[tid];
    for (int f = 0; f < IN_F; ++f)
      for (int k = 0; k < KNN; ++k)
        acc += nb[k * IN_F + f] * w1[(f * KNN + k) * E_F + tid];
    t1[tid] = lrelu(acc);
  }
  __syncthreads();
  if (tid < E_F) {
    float acc = b2[tid];
    for (int f = 0; f < E_F; ++f) acc += t1[f] * w2[f * E_F + tid];
    xf[i * XF + IN_F + tid] = lrelu(acc);
  }
}

// ---------------- K4: EdgeConv2: gather(32) -> 256x32 -> lrelu -> 32x32 -> lrelu -> xf[:,100:132] ----
__global__ void k_ec2(float* __restrict__ xf, const int* __restrict__ idx,
                      const float* __restrict__ w1, const float* __restrict__ b1,
                      const float* __restrict__ w2, const float* __restrict__ b2) {
  __shared__ float nb[KNN * E_F];
  __shared__ float t1[E_F];
  const int i = blockIdx.x, tid = threadIdx.x;
  for (int t = tid; t < KNN * E_F; t += 64) {
    const int k = t / E_F, f = t % E_F;
    nb[t] = xf[idx[i * KNN + k] * XF + IN_F + f];
  }
  __syncthreads();
  if (tid < E_F) {
    float acc = b1[tid];
    for (int f = 0; f < E_F; ++f)
      for (int k = 0; k < KNN; ++k)
        acc += nb[k * E_F + f] * w1[(f * KNN + k) * E_F + tid];
    t1[tid] = lrelu(acc);
  }
  __syncthreads();
  if (tid < E_F) {
    float acc = b2[tid];
    for (int f = 0; f < E_F; ++f) acc += t1[f] * w2[f * E_F + tid];
    xf[i * XF + IN_F + E_F + tid] = lrelu(acc);
  }
}

// -------- K5a: pair-BN stats via weighted row sums; fold into per-feature affine (a, c) --------
// left-half feature f: row i appears (n-1-i) times; right-half: row j appears j times.
__global__ void k_pstats(const float* __restrict__ xf, const float* __restrict__ lg,
                         const float* __restrict__ lb,
                         float* __restrict__ afold, float* __restrict__ cfold) {
  __shared__ double rs[128], rs2[128];
  const int f = blockIdx.x, tid = threadIdx.x;
  const int fx = (f < XF) ? f : f - XF;
  const bool left = (f < XF);
  double s = 0.0, s2 = 0.0;
  for (int r = tid; r < N_PTS; r += 128) {
    const double w = left ? (double)(N_PTS - 1 - r) : (double)r;
    const double v = (double)xf[r * XF + fx];
    s += w * v; s2 += w * v * v;
  }
  rs[tid] = s; rs2[tid] = s2; __syncthreads();
  for (int o = 64; o > 0; o >>= 1) {
    if (tid < o) { rs[tid] += rs[tid + o]; rs2[tid] += rs2[tid + o]; }
    __syncthreads();
  }
  if (tid == 0) {
    const double P = (double)NPAIR;
    const double mu = rs[0] / P;
    const double var = rs2[0] / P - mu * mu;
    const float a = lg[f] * rsqrtf((float)var + EPS);
    afold[f] = a;
    cfold[f] = lb[f] - (float)mu * a;
  }
}

// -------- K5b: d[e] = b1[e] + sum_f cfold[f] * W1[f,e] --------
__global__ void k_dvec(const float* __restrict__ cfold, const float* __restrict__ w1,
                       const float* __restrict__ b1, float* __restrict__ dvec) {
  const int e = threadIdx.x;
  float acc = b1[e];
  for (int f = 0; f < FEAT2; ++f) acc += cfold[f] * w1[f * E_F + e];
  dvec[e] = acc;
}

// -------- K5c: U = (xf*a_top) @ W1_top, V = (xf*a_bot) @ W1_bot  (each 1024x32) --------
__global__ void k_uv(const float* __restrict__ xf, const float* __restrict__ afold,
                     const float* __restrict__ w1,
                     float* __restrict__ U, float* __restrict__ V) {
  __shared__ float rowA[XF], rowB[XF];
  const int i = blockIdx.x, tid = threadIdx.x;
  for (int f = tid; f < XF; f += 64) {
    const float v = xf[i * XF + f];
    rowA[f] = v * afold[f];
    rowB[f] = v * afold[XF + f];
  }
  __syncthreads();
  const int e = tid & 31;
  if (tid < 32) {
    float acc = 0.f;
    for (int f = 0; f < XF; ++f) acc += rowA[f] * w1[f * E_F + e];
    U[i * E_F + e] = acc;
  } else {
    float acc = 0.f;
    for (int f = 0; f < XF; ++f) acc += rowB[f] * w1[(XF + f) * E_F + e];
    V[i * E_F + e] = acc;
  }
}

// -------- K6: per-pair MLP. h1 = lrelu(U[i]+V[j]+d); h2 = lrelu(h1@W2+b2) via WMMA f32;
//          out = h2@W3+b3. 128 thr = 4 waves; each wave: 4 tiles of 16 pairs. --------
__global__ void k_pair(const float* __restrict__ U, const float* __restrict__ V,
                       const float* __restrict__ dvec,
                       const float* __restrict__ w2, const float* __restrict__ b2,
                       const float* __restrict__ w3, const float* __restrict__ b3,
                       float* __restrict__ out) {
  __shared__ float h1[4][512];   // per-wave 16x32
  __shared__ float h2[4][512];
  __shared__ int   lij[4][32];   // li[0:16], lj[16:32]
  __shared__ float w3s[E_F * 2];
  __shared__ float b3s[2];
  const int tid = threadIdx.x;
  if (tid < E_F * 2) w3s[tid] = w3[tid];
  if (tid < 2) b3s[tid] = b3[tid];
  const int lane = tid & 31, wave = tid >> 5;
  const int hi = lane >> 4, col = lane & 15;

  // Preload B fragments of W2 (constant for whole kernel).
  // 16x16x4 f32 B (4x16): VGPR0 = K = kb, VGPR1 = K = kb+1, kb = 4c + 2*hi, N = col(+16*nt).
  v2f bf[8][2];
#pragma unroll
  for (int c = 0; c < 8; ++c) {
    const int kb = 4 * c + 2 * hi;
#pragma unroll
    for (int nt = 0; nt < 2; ++nt) {
      const int cg = col + 16 * nt;
      bf[c][nt].x = w2[kb * E_F + cg];
      bf[c][nt].y = w2[(kb + 1) * E_F + cg];
    }
  }
  const float dl = dvec[lane];
  const float b2lo = b2[col], b2hi = b2[col + 16];
  __syncthreads();

  const int pwave = blockIdx.x * 256 + wave * 64;
  for (int tile = 0; tile < 4; ++tile) {
    const int pbase = pwave + tile * 16;

    // decode (i,j) from flat upper-triangle index p: f(i) = i*(2n-1-i)/2
    if (lane < 16) {
      const int p = pbase + lane;
      const double disc = 2047.0 * 2047.0 - 8.0 * (double)p;
      int i = (int)((2047.0 - sqrt(disc)) * 0.5);
      if (i < 0) i = 0;
      if (i > N_PTS - 2) i = N_PTS - 2;
      while ((i + 1) * (2047 - (i + 1)) / 2 <= p) ++i;
      while (i * (2047 - i) / 2 > p) --i;
      const int j = p - i * (2047 - i) / 2 + i + 1;
      lij[wave][lane] = i;
      lij[wave][16 + lane] = j;
    }
    __syncthreads();

    // h1[t][feat=lane] = lrelu(U[i] + V[j] + d)
    for (int t = 0; t < 16; ++t) {
      const int ii = lij[wave][t], jj = lij[wave][16 + t];
      h1[wave][t * 32 + lane] = lrelu(U[ii * 32 + lane] + V[jj * 32 + lane] + dl);
    }
    __syncthreads();

    // C(16x32) = A(16x32) * W2(32x32) via 8 K-chunks x 2 N-tiles of V_WMMA_F32_16X16X4_F32.
    // A (16x4): lanes 0-15 row M=lane (K=kb,kb+1), lanes 16-31 same rows (K=kb+2,kb+3).
    v8f c0 = {}, c1 = {};
#pragma unroll
    for (int c = 0; c < 8; ++c) {
      const int kb = 4 * c + 2 * hi;
      v2f a;
      a.x = h1[wave][col * 32 + kb];
      a.y = h1[wave][col * 32 + kb + 1];
      c0 = __builtin_amdgcn_wmma_f32_16x16x4_f32(false, a, false, bf[c][0], (short)0, c0, false, false);
      c1 = __builtin_amdgcn_wmma_f32_16x16x4_f32(false, a, false, bf[c][1], (short)0, c1, false, false);
    }

    // C/D layout: VGPR r, lanes 0-15 -> (M=r, N=col); lanes 16-31 -> (M=r+8, N=col)
#pragma unroll
    for (int r = 0; r < 8; ++r) {
      const int M = r + 8 * hi;
      h2[wave][M * 32 + col]      = lrelu(c0[r] + b2lo);
      h2[wave][M * 32 + col + 16] = lrelu(c1[r] + b2hi);
    }
    __syncthreads();

    // out = h2 @ W3 + b3 : lane handles (pair=col, class=hi), 32 MACs.
    {
      float acc = b3s[hi];
      for (int f = 0; f < E_F; ++f) acc += h2[wave][col * 32 + f] * w3s[f * 2 + hi];
      out[(pbase + col) * 2 + hi] = acc;
    }
    __syncthreads();
  }
}

extern "C" void kernel_launch(void* const* d_in, const int* in_sizes, int n_in,
                              void* d_out, int out_size, void* d_ws, size_t ws_size,
                              hipStream_t stream) {
  (void)in_sizes; (void)n_in; (void)out_size; (void)ws_size;
  const float* feat   = (const float*)d_in[0];
  const float* bn_g   = (const float*)d_in[1];
  const float* bn_b   = (const float*)d_in[2];
  const float* ec1_w1 = (const float*)d_in[3];
  const float* ec1_b1 = (const float*)d_in[4];
  const float* ec1_w2 = (const float*)d_in[5];
  const float* ec1_b2 = (const float*)d_in[6];
  const float* ec2_w1 = (const float*)d_in[7];
  const float* ec2_b1 = (const float*)d_in[8];
  const float* ec2_w2 = (const float*)d_in[9];
  const float* ec2_b2 = (const float*)d_in[10];
  const float* lbn_g  = (const float*)d_in[11];
  const float* lbn_b  = (const float*)d_in[12];
  const float* lin_w1 = (const float*)d_in[13];
  const float* lin_b1 = (const float*)d_in[14];
  const float* lin_w2 = (const float*)d_in[15];
  const float* lin_b2 = (const float*)d_in[16];
  const float* lin_w3 = (const float*)d_in[17];
  const float* lin_b3 = (const float*)d_in[18];
  float* out = (float*)d_out;

  // workspace layout (floats)
  float* ws    = (float*)d_ws;
  float* xf    = ws;                   // 1024*132 = 135168
  float* U     = ws + 135168;          // 32768
  float* V     = ws + 167936;          // 32768
  float* dvec  = ws + 200704;          // 32
  float* afold = ws + 200736;          // 264
  float* cfold = ws + 201000;          // 264
  int*   idx   = (int*)(ws + 201264);  // 1024*8 ints

  k_bn    <<<IN_F, 128, 0, stream>>>(feat, bn_g, bn_b, xf);
  k_knn   <<<N_PTS, 128, 0, stream>>>(xf, idx);
  k_ec1   <<<N_PTS, 64, 0, stream>>>(xf, idx, ec1_w1, ec1_b1, ec1_w2, ec1_b2);
  k_ec2   <<<N_PTS, 64, 0, stream>>>(xf, idx, ec2_w1, ec2_b1, ec2_w2, ec2_b2);
  k_pstats<<<FEAT2, 128, 0, stream>>>(xf, lbn_g, lbn_b, afold, cfold);
  k_dvec  <<<1, 32, 0, stream>>>(cfold, lin_w1, lin_b1, dvec);
  k_uv    <<<N_PTS, 64, 0, stream>>>(xf, afold, lin_w1, U, V);
  k_pair  <<<NPAIR / 256, 128, 0, stream>>>(U, V, dvec, lin_w2, lin_b2, lin_w3, lin_b3, out);
}